// GCN_18176301596999
// MI455X (gfx1250) — compile-verified
//
#include <hip/hip_runtime.h>

// ---------------------------------------------------------------------------
// 2-layer GCN for MI455X (gfx1250, wave32).
//   agg1 = scatter_add(x[src1] -> dst1)           [N,128]
//   h    = agg1 @ W1                               [N,128]
//   t    = h @ W2                                  [N,64]
//   out  = scatter_add(t[src2] -> dst2)            [N,64]
// Memory-bound (~50 MB HBM-essential, fits L2); GEMMs are tiny (2.4 GFLOP)
// so we run them at full fp32 precision on the WMMA pipe
// (v_wmma_f32_16x16x4_f32).
// ---------------------------------------------------------------------------

typedef __attribute__((ext_vector_type(2))) float v2f;
typedef __attribute__((ext_vector_type(8))) float v8f;

// --------------------------- zero fill (float4) ----------------------------
__global__ void GCN_zero_kernel(float* __restrict__ p, int n4) {
  int i = blockIdx.x * blockDim.x + threadIdx.x;
  if (i < n4) {
    ((float4*)p)[i] = make_float4(0.f, 0.f, 0.f, 0.f);
  }
}

// ------------------------ edge-parallel scatter-add ------------------------
// F floats per node; F/4 threads per edge, each handling one float4 chunk.
// Gather load is fully coalesced; scatter uses global_atomic_add_f32.
template <int F>
__global__ void GCN_spmm_kernel(const float* __restrict__ feat,
                                const int* __restrict__ src,
                                const int* __restrict__ dst,
                                float* __restrict__ out,
                                int num_edges) {
  constexpr int TPE = F / 4;  // threads per edge
  long long gtid = (long long)blockIdx.x * blockDim.x + threadIdx.x;
  long long e = gtid / TPE;
  int j = (int)(gtid % TPE) * 4;
  if (e >= num_edges) return;
  int s = src[e];  // wave-uniform -> scalar load
  int d = dst[e];
  const float4 v = *(const float4*)(feat + (long long)s * F + j);
  float* o = out + (long long)d * F + j;
  atomicAdd(o + 0, v.x);
  atomicAdd(o + 1, v.y);
  atomicAdd(o + 2, v.z);
  atomicAdd(o + 3, v.w);
}

// ------------------------------ WMMA GEMM ----------------------------------
// D[nrows,NCOL] = A[nrows,K] * B[K,NCOL], fp32 WMMA 16x16x4.
// One block = one 16-row strip of A staged in LDS; wave w computes the
// 16x16 tile at columns [16w, 16w+16). NCOL/16 waves per block.
//
// Per ISA 7.12.2 (32-bit operands, wave32):
//   A 16x4 : lane L -> M = L&15 ; VGPR0 = K(2*(L>>4)), VGPR1 = K(2*(L>>4)+1)
//   B 4x16 : lane L -> N = L&15 ; VGPR0 = K(2*(L>>4)), VGPR1 = K(2*(L>>4)+1)
//   D 16x16: lane L -> N = L&15 ; VGPR r -> M = r + 8*(L>>4)
template <int K, int NCOL>
__global__ void GCN_gemm_wmma_kernel(const float* __restrict__ A,
                                     const float* __restrict__ B,
                                     float* __restrict__ D,
                                     int nrows) {
  constexpr int LDSROW = K + 2;  // +2 pad: 16 row-readers hit distinct banks
  __shared__ float As[16 * LDSROW];

  const int m0 = blockIdx.x * 16;
  const int lane = threadIdx.x & 31;
  const int wave = threadIdx.x >> 5;

  // Cooperatively stage the 16 x K fp32 A strip into LDS (float4 loads).
  for (int idx = threadIdx.x; idx < 16 * (K / 4); idx += blockDim.x) {
    int r = idx / (K / 4);
    int c = (idx % (K / 4)) * 4;
    float4 v = make_float4(0.f, 0.f, 0.f, 0.f);
    if (m0 + r < nrows) {
      v = *(const float4*)(A + (long long)(m0 + r) * K + c);
    }
    As[r * LDSROW + c + 0] = v.x;
    As[r * LDSROW + c + 1] = v.y;
    As[r * LDSROW + c + 2] = v.z;
    As[r * LDSROW + c + 3] = v.w;
  }
  __syncthreads();

  const int n0 = wave * 16;
  const int half = lane >> 4;  // 0: K pair {0,1}; 1: K pair {2,3}
  const int l16 = lane & 15;

  v8f c = {};
#pragma unroll
  for (int k0 = 0; k0 < K; k0 += 4) {
    const int kb = k0 + 2 * half;
    // A fragment: 8-byte LDS load, banks distinct across the 16 M-lanes.
    v2f a = *(const v2f*)&As[l16 * LDSROW + kb];
    // B fragment: W is tiny (<=64 KB) and L2/L0 resident.
    v2f b;
    b.x = B[(kb + 0) * NCOL + n0 + l16];
    b.y = B[(kb + 1) * NCOL + n0 + l16];
    c = __builtin_amdgcn_wmma_f32_16x16x4_f32(
        /*neg_a=*/false, a, /*neg_b=*/false, b,
        /*c_mod=*/(short)0, c, /*reuse_a=*/false, /*reuse_b=*/false);
  }

#pragma unroll
  for (int r = 0; r < 8; ++r) {
    int row = m0 + r + 8 * half;
    if (row < nrows) {
      D[(long long)row * NCOL + n0 + l16] = c[r];
    }
  }
}

// ------------------------------- launcher ----------------------------------
extern "C" void kernel_launch(void* const* d_in, const int* in_sizes, int n_in,
                              void* d_out, int out_size, void* d_ws, size_t ws_size,
                              hipStream_t stream) {
  const float* x   = (const float*)d_in[0];
  const float* W1  = (const float*)d_in[1];
  const float* W2  = (const float*)d_in[2];
  const int*  src1 = (const int*)d_in[3];
  const int*  dst1 = (const int*)d_in[4];
  const int*  src2 = (const int*)d_in[5];
  const int*  dst2 = (const int*)d_in[6];
  float* out = (float*)d_out;

  constexpr int IN = 128, H = 128, C = 64;
  const int N = in_sizes[0] / IN;
  const int E = in_sizes[3];

  // Workspace: agg1 [N*H] | h [N*H]; t [N*C] aliases agg1 (dead after GEMM1).
  float* agg = (float*)d_ws;
  float* h   = agg + (size_t)N * H;
  float* t   = agg;

  // Zero the atomic-accumulation targets (every call: deterministic output).
  {
    int n4 = N * H / 4;
    GCN_zero_kernel<<<(n4 + 255) / 256, 256, 0, stream>>>(agg, n4);
    int o4 = N * C / 4;
    GCN_zero_kernel<<<(o4 + 255) / 256, 256, 0, stream>>>(out, o4);
  }

  // Layer 1 aggregate: agg1 = scatter_add(x[src1] -> dst1), F=128.
  {
    long long threads = (long long)E * (IN / 4);
    int blocks = (int)((threads + 255) / 256);
    GCN_spmm_kernel<IN><<<blocks, 256, 0, stream>>>(x, src1, dst1, agg, E);
  }

  // h = agg1 @ W1   (K=128, NCOL=128 -> 8 waves / block)
  GCN_gemm_wmma_kernel<128, 128><<<(N + 15) / 16, 256, 0, stream>>>(agg, W1, h, N);

  // t = h @ W2      (K=128, NCOL=64  -> 4 waves / block)
  GCN_gemm_wmma_kernel<128, 64><<<(N + 15) / 16, 128, 0, stream>>>(h, W2, t, N);

  // Layer 2 aggregate: out = scatter_add(t[src2] -> dst2), F=64.
  {
    long long threads = (long long)E * (C / 4);
    int blocks = (int)((threads + 255) / 256);
    GCN_spmm_kernel<C><<<blocks, 256, 0, stream>>>(t, src2, dst2, out, E);
  }
}